// GCN_11338713662017
// MI455X (gfx1250) — compile-verified
//
#include <hip/hip_runtime.h>
#include <math.h>

typedef float v2f __attribute__((ext_vector_type(2)));
typedef float v8f __attribute__((ext_vector_type(8)));
typedef int   v4i __attribute__((ext_vector_type(4)));
typedef __attribute__((address_space(1))) v4i g_v4i;   // global int4
typedef __attribute__((address_space(3))) v4i l_v4i;   // LDS int4

#if __has_builtin(__builtin_amdgcn_global_load_async_to_lds_b128)
#define HAVE_ASYNC_LDS 1
#else
#define HAVE_ASYNC_LDS 0
#endif

__device__ __forceinline__ void async_wait0() {
#if __has_builtin(__builtin_amdgcn_s_wait_asynccnt)
  __builtin_amdgcn_s_wait_asynccnt(0);
#else
  asm volatile("s_wait_asynccnt 0" ::: "memory");
#endif
}

#define TPB 256

// ---------------- degree / normalization ----------------
__global__ void deg_init_k(float* deg, int n) {
  int i = blockIdx.x * blockDim.x + threadIdx.x;
  if (i < n) deg[i] = 1.0f;  // +1 self loop
}
__global__ void deg_count_k(const int* __restrict__ dst, float* deg, int ne) {
  int e = blockIdx.x * blockDim.x + threadIdx.x;
  if (e < ne) atomicAdd(&deg[dst[e]], 1.0f);
}
__global__ void deg_fin_k(float* deg, int n) {
  int i = blockIdx.x * blockDim.x + threadIdx.x;
  if (i < n) deg[i] = rsqrtf(deg[i]);
}

// ---------------- weight repack into WMMA lane order ----------------
// Bpack[(tile*K4 + k4)*32 + lane] = (v2f){ W[(4*k4+koff)  *Fout + col],
//                                          W[(4*k4+koff+1)*Fout + col] }
// col = tile*16 + (lane&15), koff = (lane>>4)*2; cols >= Fout zero-padded.
__global__ void pack_w_k(const float* __restrict__ W, float* __restrict__ Bp,
                         int k4shift, int Fout, int total) {
  int t = blockIdx.x * blockDim.x + threadIdx.x;
  if (t >= total) return;
  int lane = t & 31;
  int g    = t >> 5;                         // tile*K4 + k4
  int k4   = g & ((1 << k4shift) - 1);
  int tile = g >> k4shift;
  int l15  = lane & 15;
  int koff = (lane >> 4) << 1;
  int row  = (k4 << 2) + koff;
  int col  = (tile << 4) + l15;
  float b0 = 0.0f, b1 = 0.0f;
  if (col < Fout) {
    b0 = W[(size_t)row * Fout + col];
    b1 = W[(size_t)(row + 1) * Fout + col];
  }
  Bp[(size_t)t * 2]     = b0;
  Bp[(size_t)t * 2 + 1] = b1;
}

// ---------------- GEMM: Y[N,FOUT] = X[N,K] @ W  (f32 WMMA 16x16x4) ----------
// One block = one 16-row strip of X in LDS (staged via async global->LDS);
// wave w computes cols [16w,16w+16).
// A per ISA 7.12.2: lane row = lane%16, K-pair = (lane/16)*2.
// D: VGPR j -> row j + 8*(lane>=16), col = lane%16.
template <int K, int FOUT>
__global__ void gemm_wmma_f32_t(const float* __restrict__ X,
                                const float* __restrict__ Bpack,
                                float* __restrict__ Y, int N) {
  constexpr int LDA = K + 4;       // pad -> conflict-free strided LDS reads
  constexpr int K4  = K / 4;
  __shared__ float sA[16 * LDA];
  const int row0 = blockIdx.x << 4;
  const int tid  = threadIdx.x;

  constexpr int TOT4 = 4 * K;      // (16*K)/4 float4 chunks
  if (row0 + 16 <= N) {
#if HAVE_ASYNC_LDS
    for (int idx = tid; idx < TOT4; idx += blockDim.x) {
      int r  = idx / K4;
      int c4 = idx % K4;
      __builtin_amdgcn_global_load_async_to_lds_b128(
          (g_v4i*)(X + (size_t)(row0 + r) * K + (c4 << 2)),
          (l_v4i*)&sA[r * LDA + (c4 << 2)], 0, 0);
    }
    async_wait0();
#else
    for (int idx = tid; idx < TOT4; idx += blockDim.x) {
      int r  = idx / K4;
      int c4 = idx % K4;
      *(float4*)&sA[r * LDA + (c4 << 2)] =
          *(const float4*)(X + (size_t)(row0 + r) * K + (c4 << 2));
    }
#endif
  } else {
    for (int idx = tid; idx < TOT4; idx += blockDim.x) {
      int r  = idx / K4;
      int c4 = idx % K4;
      float4 v = {0.0f, 0.0f, 0.0f, 0.0f};
      if (row0 + r < N)
        v = *(const float4*)(X + (size_t)(row0 + r) * K + (c4 << 2));
      *(float4*)&sA[r * LDA + (c4 << 2)] = v;
    }
  }
  __syncthreads();

  const int lane = tid & 31;
  const int l15  = lane & 15;
  const int koff = (lane >> 4) << 1;
  const int wave = tid >> 5;
  const int n0   = wave << 4;
  const float* __restrict__ Ar = sA + l15 * LDA + koff;
  const v2f*   __restrict__ Bp = (const v2f*)Bpack + (size_t)wave * K4 * 32 + lane;

  v8f acc = {};
#pragma unroll
  for (int k4 = 0; k4 < K4; ++k4) {
    if ((k4 & 7) == 0)
      __builtin_prefetch(Bp + (size_t)(k4 + 16) * 32, 0, 0);  // global_prefetch_b8
    v2f a;
    a.x = Ar[(k4 << 2)];
    a.y = Ar[(k4 << 2) + 1];
    v2f b = Bp[(size_t)k4 * 32];
    acc = __builtin_amdgcn_wmma_f32_16x16x4_f32(false, a, false, b, (short)0, acc,
                                                false, false);
  }

  const int col = n0 + l15;
  if (col < FOUT) {
    const int rb = (lane >> 4) << 3;
    float* __restrict__ Yp = Y + (size_t)(row0 + rb) * FOUT + col;
    if (row0 + 16 <= N) {
#pragma unroll
      for (int j = 0; j < 8; ++j) Yp[(size_t)j * FOUT] = acc[j];
    } else {
#pragma unroll
      for (int j = 0; j < 8; ++j)
        if (row0 + rb + j < N) Yp[(size_t)j * FOUT] = acc[j];
    }
  }
}

// ---------------- aggregation ----------------
__global__ void self_loop_k(const float* __restrict__ h, const float* __restrict__ dinv,
                            float* __restrict__ agg, long long total, int fshift) {
  long long t = blockIdx.x * (long long)blockDim.x + threadIdx.x;
  if (t >= total) return;
  int i = (int)(t >> fshift);
  float di = dinv[i];
  agg[t] = h[t] * di * di;
}

// agg[dst] += h[src] * dinv[src]*dinv[dst], 4 floats per thread
__global__ void edge_scatter4_k(const int* __restrict__ src, const int* __restrict__ dst,
                                const float* __restrict__ dinv, const float* __restrict__ h,
                                float* __restrict__ agg, long long ne, int f4shift) {
  long long t = blockIdx.x * (long long)blockDim.x + threadIdx.x;
  if (t >= (ne << f4shift)) return;
  int e  = (int)(t >> f4shift);
  int f4 = (int)(t & ((1 << f4shift) - 1));
  int s = src[e], d = dst[e];
  float nrm = dinv[s] * dinv[d];
  const float4 hv = *(const float4*)(h + ((size_t)s << (f4shift + 2)) + ((size_t)f4 << 2));
  float* ap = agg + ((size_t)d << (f4shift + 2)) + ((size_t)f4 << 2);
  atomicAdd(ap + 0, hv.x * nrm);
  atomicAdd(ap + 1, hv.y * nrm);
  atomicAdd(ap + 2, hv.z * nrm);
  atomicAdd(ap + 3, hv.w * nrm);
}

// scalar fallback (F < 4)
__global__ void edge_scatter_k(const int* __restrict__ src, const int* __restrict__ dst,
                               const float* __restrict__ dinv, const float* __restrict__ h,
                               float* __restrict__ agg, long long ne, int fshift) {
  long long t = blockIdx.x * (long long)blockDim.x + threadIdx.x;
  if (t >= (ne << fshift)) return;
  int e = (int)(t >> fshift);
  int f = (int)(t & ((1LL << fshift) - 1));
  int s = src[e], d = dst[e];
  float nrm = dinv[s] * dinv[d];
  atomicAdd(&agg[((size_t)d << fshift) + f], h[((size_t)s << fshift) + f] * nrm);
}

__global__ void bias_act_k(float* __restrict__ agg, const float* __restrict__ bias,
                           long long total, int fmask, int relu) {
  long long t = blockIdx.x * (long long)blockDim.x + threadIdx.x;
  if (t >= total) return;
  float v = agg[t] + bias[t & fmask];
  agg[t] = relu ? fmaxf(v, 0.0f) : v;
}

// ---------------- pooling + log_softmax ----------------
__global__ void pool_zero_k(float* pooled, float* cnts, int G, int C) {
  int i = blockIdx.x * blockDim.x + threadIdx.x;
  if (i < G * C) pooled[i] = 0.0f;
  if (i < G) cnts[i] = 0.0f;
}
__global__ void pool_scatter_k(const float* __restrict__ h, const int* __restrict__ batch,
                               float* pooled, float* cnts, int n, int C) {
  int i = blockIdx.x * blockDim.x + threadIdx.x;
  if (i >= n) return;
  int g = batch[i];
  for (int c = 0; c < C; ++c) atomicAdd(&pooled[g * C + c], h[(size_t)i * C + c]);
  atomicAdd(&cnts[g], 1.0f);
}
__global__ void pool_final_k(const float* __restrict__ pooled, const float* __restrict__ cnts,
                             float* __restrict__ out, int G, int C) {
  int g = blockIdx.x * blockDim.x + threadIdx.x;
  if (g >= G) return;
  float inv = 1.0f / fmaxf(cnts[g], 1.0f);
  float m = -INFINITY;
  for (int c = 0; c < C; ++c) m = fmaxf(m, pooled[g * C + c] * inv);
  float s = 0.0f;
  for (int c = 0; c < C; ++c) s += __expf(pooled[g * C + c] * inv - m);
  float l = m + __logf(s);
  for (int c = 0; c < C; ++c) out[g * C + c] = pooled[g * C + c] * inv - l;
}

// ---------------- host helpers ----------------
static inline int ceildiv_ll(long long a, int b) { return (int)((a + b - 1) / b); }

static void agg_stage(const float* hout, float* aout, const float* dinv,
                      const int* src, const int* dst, const float* bias,
                      int N, long long nE, int F, int relu, hipStream_t stream) {
  int fshift = __builtin_ctz(F);
  long long totN = (long long)N << fshift;
  self_loop_k<<<ceildiv_ll(totN, TPB), TPB, 0, stream>>>(hout, dinv, aout, totN, fshift);
  if (F >= 4) {
    int f4shift = fshift - 2;
    long long tot = nE << f4shift;
    edge_scatter4_k<<<ceildiv_ll(tot, TPB), TPB, 0, stream>>>(src, dst, dinv, hout,
                                                              aout, nE, f4shift);
  } else {
    long long tot = nE << fshift;
    edge_scatter_k<<<ceildiv_ll(tot, TPB), TPB, 0, stream>>>(src, dst, dinv, hout,
                                                             aout, nE, fshift);
  }
  bias_act_k<<<ceildiv_ll(totN, TPB), TPB, 0, stream>>>(aout, bias, totN, F - 1, relu);
}

extern "C" void kernel_launch(void* const* d_in, const int* in_sizes, int n_in,
                              void* d_out, int out_size, void* d_ws, size_t ws_size,
                              hipStream_t stream) {
  const float* x   = (const float*)d_in[0];
  const float* W1  = (const float*)d_in[1];
  const float* b1  = (const float*)d_in[2];
  const float* W3  = (const float*)d_in[3];
  const float* b3  = (const float*)d_in[4];
  const float* W4  = (const float*)d_in[5];
  const float* b4  = (const float*)d_in[6];
  const int*   eix = (const int*)d_in[7];
  const int*   bat = (const int*)d_in[8];

  const int H1  = in_sizes[2];                 // 128
  const int H2  = in_sizes[4];                 // 32
  const int H3  = in_sizes[6];                 // 2
  const int FIN = in_sizes[1] / H1;            // 512
  const int N   = in_sizes[0] / FIN;
  const long long nE = in_sizes[7] / 2;
  const int G   = out_size / H3;
  const int* src = eix;
  const int* dst = eix + nE;

  float* ws     = (float*)d_ws;
  float* dinv   = ws;                                  // N
  float* hbuf   = dinv + N;                            // N*H1
  float* aggbuf = hbuf + (size_t)N * H1;               // N*H1
  float* pooled = aggbuf + (size_t)N * H1;             // G*H3
  float* cnts   = pooled + (size_t)G * H3;             // G
  float* bp1    = cnts + G;
  float* bp2    = bp1 + (size_t)((H1 + 15) / 16) * FIN * 16;
  float* bp3    = bp2 + (size_t)((H2 + 15) / 16) * H1 * 16;

  // degrees -> dinv
  deg_init_k<<<(N + TPB - 1) / TPB, TPB, 0, stream>>>(dinv, N);
  deg_count_k<<<ceildiv_ll(nE, TPB), TPB, 0, stream>>>(dst, dinv, (int)nE);
  deg_fin_k<<<(N + TPB - 1) / TPB, TPB, 0, stream>>>(dinv, N);

  // pack weights into WMMA lane order (cols padded to 16 with zeros)
  {
    int t1 = ((H1 + 15) / 16) * (FIN / 4) * 32;
    pack_w_k<<<(t1 + TPB - 1) / TPB, TPB, 0, stream>>>(W1, bp1,
                                                       __builtin_ctz(FIN / 4), H1, t1);
    int t2 = ((H2 + 15) / 16) * (H1 / 4) * 32;
    pack_w_k<<<(t2 + TPB - 1) / TPB, TPB, 0, stream>>>(W3, bp2,
                                                       __builtin_ctz(H1 / 4), H2, t2);
    int t3 = ((H3 + 15) / 16) * (H2 / 4) * 32;
    pack_w_k<<<(t3 + TPB - 1) / TPB, TPB, 0, stream>>>(W4, bp3,
                                                       __builtin_ctz(H2 / 4), H3, t3);
  }

  const int nstrips = (N + 15) / 16;

  // layer 1: [N,512] @ [512,128], relu
  gemm_wmma_f32_t<512, 128><<<nstrips, 32 * 8, 0, stream>>>(x, bp1, hbuf, N);
  agg_stage(hbuf, aggbuf, dinv, src, dst, b1, N, nE, 128, 1, stream);

  // layer 2: [N,128] @ [128,32], relu
  gemm_wmma_f32_t<128, 32><<<nstrips, 32 * 2, 0, stream>>>(aggbuf, bp2, hbuf, N);
  agg_stage(hbuf, aggbuf, dinv, src, dst, b3, N, nE, 32, 1, stream);

  // layer 3: [N,32] @ [32,2]
  gemm_wmma_f32_t<32, 2><<<nstrips, 32 * 1, 0, stream>>>(aggbuf, bp3, hbuf, N);
  agg_stage(hbuf, aggbuf, dinv, src, dst, b4, N, nE, 2, 0, stream);

  // pooling + log_softmax
  pool_zero_k<<<(G * H3 + TPB - 1) / TPB, TPB, 0, stream>>>(pooled, cnts, G, H3);
  pool_scatter_k<<<(N + TPB - 1) / TPB, TPB, 0, stream>>>(aggbuf, bat, pooled, cnts, N, H3);
  pool_final_k<<<(G + TPB - 1) / TPB, TPB, 0, stream>>>(pooled, cnts, (float*)d_out, G, H3);
}